// GSICell_57269093925257
// MI455X (gfx1250) — compile-verified
//
#include <hip/hip_runtime.h>
#include <stdint.h>

// Problem constants (fixed by the reference).
#define NCL     38
#define NFL     12
#define NATOMS  100
#define NTOT    10000

// ---------------- fast math helpers (native HW ops) ----------------
__device__ __forceinline__ float rcp_f(float x) { return __builtin_amdgcn_rcpf(x); }

__device__ __forceinline__ float safe_inv(float v) {
  // jnp.where(v>=0, 1, -1); 1/where(|v|<eps, s*eps, v)
  float sgn = (v >= 0.0f) ? 0.01f : -0.01f;
  float den = (__builtin_fabsf(v) < 0.01f) ? sgn : v;
  return rcp_f(den);
}
__device__ __forceinline__ float sigm_f(float v) { return rcp_f(1.0f + __expf(-v)); }
__device__ __forceinline__ float tanh_f(float v) {
  float t = __expf(-2.0f * v);
  return (1.0f - t) * rcp_f(1.0f + t);
}
__device__ __forceinline__ float relu_f(float v) { return fmaxf(v, 0.0f); }

// ---------------- TDM: DMA a 1-D f32 array (<=65535 elems) into LDS ----------------
#if defined(__has_builtin)
#if __has_builtin(__builtin_amdgcn_tensor_load_to_lds)
#define HAVE_TDM 1
#endif
#endif

#ifdef HAVE_TDM
typedef unsigned int u32x4 __attribute__((ext_vector_type(4)));
typedef int          i32x4 __attribute__((ext_vector_type(4)));
typedef int          i32x8 __attribute__((ext_vector_type(8)));

__device__ __forceinline__ void tdm_load_1d_f32(const float* gsrc, uint32_t lds_byte_off,
                                                unsigned nelem) {
  uint64_t ga = (uint64_t)(uintptr_t)gsrc;
  u32x4 g0;
  g0.x = 1u;                                            // count=1, user descriptor
  g0.y = lds_byte_off;                                  // lds_addr (bytes)
  g0.z = (uint32_t)(ga & 0xFFFFFFFFu);                  // global_addr[31:0]
  g0.w = (uint32_t)((ga >> 32) & 0x01FFFFFFu) | (2u << 30); // addr[56:32] | type=2
  i32x8 g1;
  g1[0] = (int)(2u << 16);                              // data_size=4B; wg_mask=0
  g1[1] = (int)((nelem & 0xFFFFu) << 16);               // tensor_dim0[15:0]
  g1[2] = (int)(((nelem >> 16) & 0xFFFFu) | (1u << 16));// dim0[31:16] | tensor_dim1=1
  g1[3] = (int)((nelem & 0xFFFFu) << 16);               // tile_dim0 = nelem
  g1[4] = 1;                                            // tile_dim1=1, tile_dim2=0
  g1[5] = (int)nelem;                                   // tensor_dim0_stride[31:0]
  g1[6] = 0;
  g1[7] = 0;
  i32x4 g2 = {0, 0, 0, 0};                              // dims >=3 unused (tile_dim3=0)
  i32x4 g3 = {0, 0, 0, 0};
#if __clang_major__ >= 23
  i32x8 g4 = {0, 0, 0, 0, 0, 0, 0, 0};
  __builtin_amdgcn_tensor_load_to_lds(g0, g1, g2, g3, g4, 0);
#else
  __builtin_amdgcn_tensor_load_to_lds(g0, g1, g2, g3, 0);
#endif
}

__device__ __forceinline__ void wait_tensorcnt0() {
#if __has_builtin(__builtin_amdgcn_s_wait_tensorcnt)
  __builtin_amdgcn_s_wait_tensorcnt(0);
#else
  asm volatile("s_wait_tensorcnt 0x0" ::: "memory");
#endif
}
#endif // HAVE_TDM

// ---------------- node kernel: out[n] = F(n) (F_COEF = 1) ----------------
__global__ __launch_bounds__(256) void gsic_node_kernel(
    const float* __restrict__ x, const float* __restrict__ f_mask,
    const float* __restrict__ wf2, float* __restrict__ out, int n_total) {
  int n = blockIdx.x * blockDim.x + threadIdx.x;
  if (n >= n_total) return;

  float w[NFL];
#pragma unroll
  for (int k = 0; k < NFL; ++k) w[k] = f_mask[k] * (wf2[k] - wf2[k + NFL]);

  float v  = x[n];
  float sv = __sinf(v), cv = __cosf(v);
  float iv = safe_inv(v);

  float F = w[0];                       // ones
  F += w[1] * sv + w[2] * cv + w[3] * v * sv;
  F += w[4] * v + w[5] * v * v;         // x^1, x^2
  F += w[6] * iv + w[7] * iv * iv + w[8] * iv * iv * iv;
  F += w[9] * tanh_f(v) + w[10] * sigm_f(v) + w[11] * relu_f(v);
  out[n] = F;
}

// ---------------- edge kernel: out[dst] += edge_attr * (feats . w) ----------------
__global__ __launch_bounds__(256) void gsic_edge_kernel(
    const float* __restrict__ x, const int* __restrict__ eidx,
    const float* __restrict__ c_mask, const float* __restrict__ wc2,
    const float* __restrict__ ea, float* __restrict__ out, int n_edges) {
  // 40KB + 40KB gather tables in LDS (320KB/WGP -> up to 4 blocks/WGP).
  __shared__ float lx[NTOT];
  __shared__ float lea[NTOT];

#ifdef HAVE_TDM
  if (threadIdx.x < 32) {  // wave 0 drives the DMA; TDM ignores EXEC
    tdm_load_1d_f32(x,  (uint32_t)(uintptr_t)(void*)lx,  NTOT);
    tdm_load_1d_f32(ea, (uint32_t)(uintptr_t)(void*)lea, NTOT);
    wait_tensorcnt0();
  }
  __syncthreads();
#else
  for (int i = threadIdx.x; i < NTOT; i += blockDim.x) {
    lx[i]  = x[i];
    lea[i] = ea[i];
  }
  __syncthreads();
#endif

  // Effective weights for the [C, -C] concat: w = mask * (wc2_lo - wc2_hi).
  float w[NCL];
#pragma unroll
  for (int k = 0; k < NCL; ++k) w[k] = c_mask[k] * (wc2[k] - wc2[k + NCL]);

  const int* __restrict__ srcp = eidx;            // edge_index[0] = src (x_j)
  const int* __restrict__ dstp = eidx + n_edges;  // edge_index[1] = dst (x_i)
  const int stride = gridDim.x * blockDim.x;

  for (int e = blockIdx.x * blockDim.x + threadIdx.x; e < n_edges; e += stride) {
    __builtin_prefetch(srcp + e + stride, 0, 0);  // global_prefetch_b8 on the streams
    __builtin_prefetch(dstp + e + stride, 0, 0);

    uint32_t s = (uint32_t)srcp[e];
    uint32_t d = (uint32_t)dstp[e];
    float xj = lx[s];   // x_j = x[src]
    float xi = lx[d];   // x_i = x[dst]
    float dd = xj - xi;
    float sm = xj + xi;

    float acc = w[0] * xi + w[1] * xj + w[2] * dd;          // p=1
    acc += w[3] * xi * xi + w[4] * xj * xj + w[5] * dd * dd; // p=2

    float ixi = safe_inv(xi), ixj = safe_inv(xj);
    float idd = safe_inv(dd), ism = safe_inv(sm);
    acc += w[6] * ixi + w[7] * ixj + w[8] * idd + w[9] * ism;                     // n=1
    acc += w[10] * ixi * ixi + w[11] * ixj * ixj + w[12] * idd * idd + w[13] * ism * ism; // n=2

    float sd = __sinf(dd), cd = __cosf(dd);
    float ss = __sinf(sm), cs = __cosf(sm);
    float si = __sinf(xi), ci = __cosf(xi);
    float sj = __sinf(xj), cj = __cosf(xj);
    acc += w[14] * sd + w[15] * cd + w[16] * ss + w[17] * cs;
    acc += w[18] * si + w[19] * ci + w[20] * sj + w[21] * cj;
    acc += w[22] * xi * xj + w[23] * xi * sj + w[24] * xj * si + w[25] * sd * cs;

    acc += w[26] * tanh_f(xi) + w[27] * sigm_f(xi) + w[28] * relu_f(xi);
    acc += w[29] * tanh_f(xj) + w[30] * sigm_f(xj) + w[31] * relu_f(xj);
    acc += w[32] * tanh_f(dd) + w[33] * sigm_f(dd) + w[34] * relu_f(dd);
    acc += w[35] * tanh_f(sm) + w[36] * sigm_f(sm) + w[37] * relu_f(sm);

    float eav = lea[(s % NATOMS) * NATOMS + (d % NATOMS)];
    unsafeAtomicAdd(out + d, eav * acc);  // global_atomic_add_f32 (no return)
  }
}

// ---------------- host launcher ----------------
// d_in order: t, x, edge_index, c_mask, f_mask, wc_2, wf_2, edge_attr_all
extern "C" void kernel_launch(void* const* d_in, const int* in_sizes, int n_in,
                              void* d_out, int out_size, void* d_ws, size_t ws_size,
                              hipStream_t stream) {
  (void)n_in; (void)d_ws; (void)ws_size; (void)out_size;
  const float* x      = (const float*)d_in[1];
  const int*   eidx   = (const int*)d_in[2];   // harness convention: integer -> int32
  const float* c_mask = (const float*)d_in[3];
  const float* f_mask = (const float*)d_in[4];
  const float* wc2    = (const float*)d_in[5];
  const float* wf2    = (const float*)d_in[6];
  const float* ea     = (const float*)d_in[7];
  float* out = (float*)d_out;

  const int n_total = in_sizes[1];       // 10000
  const int n_edges = in_sizes[2] / 2;   // 1,000,000

  // 1) out = F (overwrites 0xAA poison), 2) edge scatter-adds on top. Same
  // stream => ordered under graph capture.
  gsic_node_kernel<<<(n_total + 255) / 256, 256, 0, stream>>>(x, f_mask, wf2, out, n_total);
  gsic_edge_kernel<<<256, 256, 0, stream>>>(x, eidx, c_mask, wc2, ea, out, n_edges);
}